// AttributeDecoder_39865886442297
// MI455X (gfx1250) — compile-verified
//
#include <hip/hip_runtime.h>

typedef __attribute__((ext_vector_type(2))) float v2f;
typedef __attribute__((ext_vector_type(8))) float v8f;

// ---------------------------------------------------------------------------
// Degree / normalization kernels
// ---------------------------------------------------------------------------
__global__ void gcn_init_deg(float* deg, int n) {
    int i = blockIdx.x * blockDim.x + threadIdx.x;
    if (i < n) deg[i] = 1.0f;               // self-loop contributes 1
}

__global__ void gcn_deg_accum(float* deg, const long long* __restrict__ dst, long long E) {
    long long e = (long long)blockIdx.x * blockDim.x + threadIdx.x;
    if (e < E) atomicAdd(&deg[(int)dst[e]], 1.0f);
}

__global__ void gcn_deg_to_dinv(float* deg, int n) {
    int i = blockIdx.x * blockDim.x + threadIdx.x;
    if (i < n) {
        float d = deg[i];                   // always >= 1 (self-loop)
        deg[i] = rsqrtf(d);
    }
}

// ---------------------------------------------------------------------------
// Dense GEMM C[n,M] = A[n,K] @ B[K,M] using V_WMMA_F32_16X16X4_F32.
// One wave computes one 16x16 tile of C. 4 waves per block.
//   A 16x4 tile layout : lanes 0-15 = rows, VGPR0/1 = K+0,K+1 ; lanes 16-31 = K+2,K+3
//   B 4x16 tile layout : lanes 0-15 = cols, VGPR0/1 = K+0,K+1 ; lanes 16-31 = K+2,K+3
//   C 16x16 layout     : VGPR j = row j (lanes 0-15) / row j+8 (lanes 16-31), col = lane&15
// Requires n % 16 == 0, K % 4 == 0, M % 16 == 0 (holds: 50000, 64/128, 128/64).
// ---------------------------------------------------------------------------
template <int K, int M>
__global__ void gcn_gemm_wmma(const float* __restrict__ A,
                              const float* __restrict__ B,
                              float* __restrict__ C) {
    const int lane = threadIdx.x & 31;
    const int wave = threadIdx.x >> 5;
    const int half = lane >> 4;             // 0: lanes 0-15, 1: lanes 16-31
    const int l    = lane & 15;
    const int row0 = blockIdx.x * 16;
    const int col0 = (blockIdx.y * 4 + wave) * 16;

    const float* __restrict__ Arow = A + (size_t)(row0 + l) * K;

    v8f acc = {};
#pragma unroll
    for (int k = 0; k < K; k += 4) {
        const int ka = k + 2 * half;
        v2f a, b;
        a.x = Arow[ka];
        a.y = Arow[ka + 1];
        b.x = B[(size_t)ka * M + col0 + l];
        b.y = B[(size_t)(ka + 1) * M + col0 + l];
        acc = __builtin_amdgcn_wmma_f32_16x16x4_f32(
            /*neg_a=*/false, a, /*neg_b=*/false, b,
            /*c_mod=*/(short)0, acc, /*reuse_a=*/false, /*reuse_b=*/false);
    }

#pragma unroll
    for (int j = 0; j < 8; ++j) {
        C[(size_t)(row0 + j + 8 * half) * M + col0 + l] = acc[j];
    }
}

// ---------------------------------------------------------------------------
// Scatter-add with symmetric normalization. Edges e in [0,E) come from the
// edge list; e in [E, E+n) are self-loops (src=dst=e-E). Each thread handles
// one float4 chunk of one edge's feature row.  F = 4 << logF4.
// ---------------------------------------------------------------------------
__global__ void gcn_scatter_add(const float* __restrict__ h,
                                const float* __restrict__ dinv,
                                const long long* __restrict__ src,
                                const long long* __restrict__ dst,
                                float* __restrict__ out,
                                long long E, int n, int logF4) {
    const long long t = (long long)blockIdx.x * blockDim.x + threadIdx.x;
    const long long total = ((long long)E + n) << logF4;
    if (t >= total) return;

    const long long e = t >> logF4;
    const int c = (int)(t & ((1 << logF4) - 1));
    int s, d;
    if (e < E) {
        s = (int)src[e];
        d = (int)dst[e];
    } else {
        s = d = (int)(e - E);
    }
    const float nrm = dinv[s] * dinv[d];
    const int F = 4 << logF4;

    const float4 v = *reinterpret_cast<const float4*>(h + (size_t)s * F + (size_t)c * 4);
    float* o = out + (size_t)d * F + (size_t)c * 4;
    atomicAdd(o + 0, v.x * nrm);
    atomicAdd(o + 1, v.y * nrm);
    atomicAdd(o + 2, v.z * nrm);
    atomicAdd(o + 3, v.w * nrm);
}

// ---------------------------------------------------------------------------
// out[i,f] = relu(out[i,f] + bias[f]) in place.  F is a power of two.
// ---------------------------------------------------------------------------
__global__ void gcn_bias_relu(float* __restrict__ x, const float* __restrict__ b,
                              long long total, int Fmask) {
    long long t = (long long)blockIdx.x * blockDim.x + threadIdx.x;
    if (t >= total) return;
    float v = x[t] + b[(int)(t & Fmask)];
    x[t] = v > 0.0f ? v : 0.0f;
}

// ---------------------------------------------------------------------------
extern "C" void kernel_launch(void* const* d_in, const int* in_sizes, int n_in,
                              void* d_out, int out_size, void* d_ws, size_t ws_size,
                              hipStream_t stream) {
    const float*     z  = (const float*)d_in[0];
    const long long* ei = (const long long*)d_in[1];   // int64 edge_index [2,E]
    const float*     W1 = (const float*)d_in[2];       // [64,128]
    const float*     b1 = (const float*)d_in[3];       // [128]
    const float*     W2 = (const float*)d_in[4];       // [128,64]
    const float*     b2 = (const float*)d_in[5];       // [64]

    const int n       = in_sizes[0] / 64;              // 50000 (multiple of 16)
    const long long E = (long long)in_sizes[1] / 2;    // 1.6M
    const long long* src = ei;
    const long long* dst = ei + E;

    // Workspace layout: dinv[n] | h[n*128] (reused for h2[n*64]) | agg1[n*128]
    char* ws = (char*)d_ws;
    float* dinv = (float*)ws;
    size_t off = ((size_t)n * sizeof(float) + 255) & ~(size_t)255;
    float* h = (float*)(ws + off);
    off += ((size_t)n * 128 * sizeof(float) + 255) & ~(size_t)255;
    float* agg1 = (float*)(ws + off);

    float* out = (float*)d_out;                        // [n,64]

    hipMemsetAsync(agg1, 0, (size_t)n * 128 * sizeof(float), stream);
    hipMemsetAsync(out, 0, (size_t)n * 64 * sizeof(float), stream);

    const int B = 256;

    // --- degrees -> dinv (in place) ---
    gcn_init_deg<<<(n + B - 1) / B, B, 0, stream>>>(dinv, n);
    gcn_deg_accum<<<(unsigned)((E + B - 1) / B), B, 0, stream>>>(dinv, dst, E);
    gcn_deg_to_dinv<<<(n + B - 1) / B, B, 0, stream>>>(dinv, n);

    // --- layer 1: h1 = z @ W1 ; agg1 = scatter(norm * h1[src]) ; relu(+b1) ---
    gcn_gemm_wmma<64, 128><<<dim3(n / 16, 2), 128, 0, stream>>>(z, W1, h);

    {
        long long total = ((long long)E + n) << 5;     // F=128 -> 32 float4 chunks
        gcn_scatter_add<<<(unsigned)((total + B - 1) / B), B, 0, stream>>>(
            h, dinv, src, dst, agg1, E, n, 5);
        long long tot2 = (long long)n * 128;
        gcn_bias_relu<<<(unsigned)((tot2 + B - 1) / B), B, 0, stream>>>(agg1, b1, tot2, 127);
    }

    // --- layer 2: h2 = agg1 @ W2 (reuse h buffer) ; out = scatter ; relu(+b2) ---
    gcn_gemm_wmma<128, 64><<<dim3(n / 16, 1), 128, 0, stream>>>(agg1, W2, h);

    {
        long long total = ((long long)E + n) << 4;     // F=64 -> 16 float4 chunks
        gcn_scatter_add<<<(unsigned)((total + B - 1) / B), B, 0, stream>>>(
            h, dinv, src, dst, out, E, n, 4);
        long long tot2 = (long long)n * 64;
        gcn_bias_relu<<<(unsigned)((tot2 + B - 1) / B), B, 0, stream>>>(out, b2, tot2, 63);
    }
}